// DepthAugmentation_37391985279486
// MI455X (gfx1250) — compile-verified
//
#include <hip/hip_runtime.h>
#include <stdint.h>

// Problem constants (from the reference)
#define BATCH 64
#define HH 480
#define WW 640
#define HS (HH / 4)   // 120
#define WS (WW / 4)   // 160
#define NOISE_SIGMA 0.005f
#define P_DROPOUT 0.003125f
#define P_RANDOM 0.003125f
#define P_STICK 0.00025f

typedef float v4f __attribute__((ext_vector_type(4)));

// gfx1250 async global->LDS copy path (ASYNCcnt), guarded so the file always compiles.
#if defined(__HIP_DEVICE_COMPILE__) &&                                          \
    __has_builtin(__builtin_amdgcn_global_load_async_to_lds_b32) &&             \
    __has_builtin(__builtin_amdgcn_s_wait_asynccnt)
#define USE_ASYNC_LDS 1
#else
#define USE_ASYNC_LDS 0
#endif

#if USE_ASYNC_LDS
typedef __attribute__((address_space(1))) int  gint_t;
typedef __attribute__((address_space(3))) int  lint_t;
#endif

__device__ __forceinline__ float clamp01(float v) {
    return fminf(fmaxf(v, 0.0f), 1.0f);
}

// Bilinear sample of the (HS,WS) noise grid at output pixel (y,x), half-pixel
// centers with edge clamp (matches jax.image.resize 'bilinear' at 4x upscale).
__device__ __forceinline__ float bilerp_noise_global(const float* __restrict__ nb,
                                                     int y, int x) {
    float fy = y * 0.25f - 0.375f;
    fy = fminf(fmaxf(fy, 0.0f), (float)(HS - 1));
    int   y0 = (int)fy;
    float wy = fy - (float)y0;
    int   y1 = min(y0 + 1, HS - 1);

    float fx = x * 0.25f - 0.375f;
    fx = fminf(fmaxf(fx, 0.0f), (float)(WS - 1));
    int   x0 = (int)fx;
    float wx = fx - (float)x0;
    int   x1 = min(x0 + 1, WS - 1);

    float a = nb[y0 * WS + x0], b = nb[y0 * WS + x1];
    float c = nb[y1 * WS + x0], d = nb[y1 * WS + x1];
    float t0 = a + wx * (b - a);
    float t1 = c + wx * (d - c);
    return t0 + wy * (t1 - t0);
}

// The scalar augmentation applied to a single pixel (matches kernel 2 exactly).
__device__ __forceinline__ float augment_pixel(float d, float noise, float du,
                                               float ru, float rv) {
    float nv = (d > 0.0f) ? noise * NOISE_SIGMA : 0.0f;
    d = clamp01(d + nv);
    d = (du >= P_DROPOUT) ? d : 0.0f;
    d = (ru < P_RANDOM) ? rv : d;
    return d;
}

// ---------------------------------------------------------------------------
// Kernel 1: per-batch stick parameters + stick value (depends on the
// *augmented* depth at the anchor pixel). Writes 8 ints per batch to d_ws:
// [y0, y1, x0, x1, val_bits, on, 0, 0]
// ---------------------------------------------------------------------------
__global__ __launch_bounds__(64) void DepthAug_stick_prep(
    const float* __restrict__ depth, const float* __restrict__ noise_lo,
    const float* __restrict__ dropout_u, const float* __restrict__ random_u,
    const float* __restrict__ random_vals, const float* __restrict__ stick_u,
    const float* __restrict__ horiz_u, const float* __restrict__ fallback_vals,
    const int* __restrict__ stick_len, const int* __restrict__ stick_width,
    const int* __restrict__ stick_y, const int* __restrict__ stick_x,
    int* __restrict__ ws) {
    int b = threadIdx.x;
    if (b >= BATCH) return;

    int  on    = (stick_u[b] < P_STICK * (float)HH * (float)WW) ? 1 : 0;
    bool horiz = horiz_u[b] > 0.5f;
    int  len   = stick_len[b] + 1;
    int  wid   = stick_width[b] + 1;
    int  sh    = horiz ? wid : len;
    int  sw    = horiz ? len : wid;
    int  y     = min(max(stick_y[b], 0), max(HH - sh, 1) - 1);
    int  x     = min(max(stick_x[b], 0), max(WW - sw, 1) - 1);

    size_t idx = ((size_t)b * HH + (size_t)y) * WW + (size_t)x;
    const float* nb = noise_lo + (size_t)b * HS * WS;

    float d = augment_pixel(depth[idx], bilerp_noise_global(nb, y, x),
                            dropout_u[idx], random_u[idx], random_vals[idx]);
    float val = (d > 0.0f) ? d : fallback_vals[b];

    int* w8 = ws + b * 8;
    w8[0] = y;
    w8[1] = y + sh;
    w8[2] = x;
    w8[3] = x + sw;
    w8[4] = __float_as_int(val);
    w8[5] = on;
    w8[6] = 0;
    w8[7] = 0;
}

// ---------------------------------------------------------------------------
// Kernel 2: one block per (batch, row). 160 threads (5 wave32) * 4 px/lane.
// Noise rows staged to LDS via gfx1250 async-to-LDS DMA; all bulk streams use
// non-temporal B128 VMEM so the reused noise grid stays resident in L2.
// ---------------------------------------------------------------------------
__global__ __launch_bounds__(WS) void DepthAug_main(
    const float* __restrict__ depth, const float* __restrict__ noise_lo,
    const float* __restrict__ dropout_u, const float* __restrict__ random_u,
    const float* __restrict__ random_vals, const int* __restrict__ ws,
    float* __restrict__ out) {
    __shared__ float sN[2 * WS];

    const int yy = blockIdx.x;  // row      [0, HH)
    const int b  = blockIdx.y;  // batch    [0, BATCH)
    const int t  = threadIdx.x; // lane job [0, WS): one float4 of the row

    // Vertical bilinear coordinates (uniform across the block).
    float fy = yy * 0.25f - 0.375f;
    fy = fminf(fmaxf(fy, 0.0f), (float)(HS - 1));
    const int   y0 = (int)fy;
    const float wy = fy - (float)y0;
    const int   y1 = min(y0 + 1, HS - 1);

    const float* nb = noise_lo + (size_t)b * HS * WS;

#if USE_ASYNC_LDS
    {
        const float* g0 = nb + y0 * WS + t;
        const float* g1 = nb + y1 * WS + t;
        __builtin_amdgcn_global_load_async_to_lds_b32(
            (gint_t*)(uintptr_t)g0,
            (lint_t*)(unsigned)(uintptr_t)(&sN[t]),
            0, 0);
        __builtin_amdgcn_global_load_async_to_lds_b32(
            (gint_t*)(uintptr_t)g1,
            (lint_t*)(unsigned)(uintptr_t)(&sN[WS + t]),
            0, 0);
        __builtin_amdgcn_s_wait_asynccnt(0);
    }
    __syncthreads();
#else
    sN[t]      = nb[y0 * WS + t];
    sN[WS + t] = nb[y1 * WS + t];
    __syncthreads();
#endif

    const size_t rowBase = ((size_t)b * HH + (size_t)yy) * WW;

    // Prime L2 two rows ahead for the four streaming inputs (global_prefetch).
    if (yy + 2 < HH) {
        const size_t pf = rowBase + (size_t)2 * WW + (size_t)4 * t;
        __builtin_prefetch(depth + pf, 0, 0);
        __builtin_prefetch(dropout_u + pf, 0, 0);
        __builtin_prefetch(random_u + pf, 0, 0);
        __builtin_prefetch(random_vals + pf, 0, 0);
    }

    // Single-use streams: non-temporal B128 loads.
    v4f d4  = __builtin_nontemporal_load(((const v4f*)(depth + rowBase)) + t);
    v4f du4 = __builtin_nontemporal_load(((const v4f*)(dropout_u + rowBase)) + t);
    v4f ru4 = __builtin_nontemporal_load(((const v4f*)(random_u + rowBase)) + t);
    v4f rv4 = __builtin_nontemporal_load(((const v4f*)(random_vals + rowBase)) + t);

    // Per-batch stick parameters (uniform across the block -> scalar loads).
    const int*  w8   = ws + b * 8;
    const int   sy0  = w8[0], sy1 = w8[1], sx0 = w8[2], sx1 = w8[3];
    const float sval = __int_as_float(w8[4]);
    const bool rowin = (w8[5] != 0) && (yy >= sy0) && (yy < sy1);

    v4f o4;
#pragma unroll
    for (int j = 0; j < 4; ++j) {
        const int x = 4 * t + j;

        // Horizontal bilinear from the LDS-staged rows.
        float fx = x * 0.25f - 0.375f;
        fx = fminf(fmaxf(fx, 0.0f), (float)(WS - 1));
        const int   x0 = (int)fx;
        const float wx = fx - (float)x0;
        const int   x1 = min(x0 + 1, WS - 1);

        const float n0 = sN[x0] + wx * (sN[x1] - sN[x0]);
        const float n1 = sN[WS + x0] + wx * (sN[WS + x1] - sN[WS + x0]);
        const float nraw = n0 + wy * (n1 - n0);

        float d = augment_pixel(d4[j], nraw, du4[j], ru4[j], rv4[j]);
        if (rowin && x >= sx0 && x < sx1) d = sval;
        o4[j] = d;
    }

    __builtin_nontemporal_store(o4, ((v4f*)(out + rowBase)) + t);
}

// ---------------------------------------------------------------------------
extern "C" void kernel_launch(void* const* d_in, const int* in_sizes, int n_in,
                              void* d_out, int out_size, void* d_ws, size_t ws_size,
                              hipStream_t stream) {
    (void)in_sizes; (void)n_in; (void)out_size; (void)ws_size;

    const float* depth         = (const float*)d_in[0];
    const float* noise_lo      = (const float*)d_in[1];
    const float* dropout_u     = (const float*)d_in[2];
    const float* random_u      = (const float*)d_in[3];
    const float* random_vals   = (const float*)d_in[4];
    const float* stick_u       = (const float*)d_in[5];
    const float* horiz_u       = (const float*)d_in[6];
    const float* fallback_vals = (const float*)d_in[7];
    const int*   stick_len     = (const int*)d_in[8];
    const int*   stick_width   = (const int*)d_in[9];
    const int*   stick_y       = (const int*)d_in[10];
    const int*   stick_x       = (const int*)d_in[11];

    float* out = (float*)d_out;
    int*   ws  = (int*)d_ws;

    hipLaunchKernelGGL(DepthAug_stick_prep, dim3(1), dim3(64), 0, stream,
                       depth, noise_lo, dropout_u, random_u, random_vals,
                       stick_u, horiz_u, fallback_vals,
                       stick_len, stick_width, stick_y, stick_x, ws);

    hipLaunchKernelGGL(DepthAug_main, dim3(HH, BATCH), dim3(WS), 0, stream,
                       depth, noise_lo, dropout_u, random_u, random_vals, ws, out);
}